// Patcher_45758581571826
// MI455X (gfx1250) — compile-verified
//
#include <hip/hip_runtime.h>
#include <stdint.h>

// ============================================================================
// Patcher (unfold 64x25/stride 64x12 + [21248x1600]x[1600x256] GEMM + GELU)
// for MI455X (gfx1250).
//
//  * fp32 compute-bound (AI ~195 FLOP/B) -> V_WMMA_F32_16X16X4_F32.
//  * Unfold by Tensor Data Mover iterate mode, split into TWO descriptors
//    (x rows 0..31 -> k in [0,800), rows 32..63 -> k in [800,1600)) so the
//    second DMA overlaps with compute on the first K-half:
//        issue TDM0+TDM1 ; s_wait_tensorcnt<=1 ; barrier ;
//        K-loop [0,800)  ; s_wait_tensorcnt==0 ; barrier ;
//        K-loop [800,1600)
//  * Full K (1600) resident in LDS (205 KB of 320 KB WGP LDS), +4-float
//    per-patch pad => conflict-free ds_load_b64 A-fragments.
//  * Block: M=32 patches x N=256, 8 waves; wave = 32x32 outputs
//    (4 f32 accumulators, 4 WMMAs per K-step of 4).
// ============================================================================

typedef __attribute__((ext_vector_type(2))) float        v2f;
typedef __attribute__((ext_vector_type(8))) float        v8f;
typedef __attribute__((ext_vector_type(4))) unsigned int v4u;
typedef __attribute__((ext_vector_type(4))) int          v4i;
typedef __attribute__((ext_vector_type(8))) int          v8i;

#define XW      4000            // x row length
#define PH      64              // patch height == x height (nh = 1)
#define PW      25              // patch width
#define SW      12              // patch stride (width)
#define NPATCH  332             // patches per batch image
#define NBATCH  64
#define KDIM    1600            // PH*PW
#define KHALF   800             // rows 0..31 of each patch
#define ROWSPLIT 32             // x rows per TDM descriptor
#define NDIM    256             // output features
#define MTILE   32              // patches per block
#define APAD    4               // LDS pad (floats) -> conflict-free b64 reads
#define ASTRIDE (KDIM + APAD)   // 1604 floats per patch in LDS

__device__ __forceinline__ float gelu_exact(float v) {
    // exact GELU: 0.5*x*(1+erf(x/sqrt(2))), matching approximate=False
    return 0.5f * v * (1.0f + erff(v * 0.70710678118654752440f));
}

// Issue one iterated 2D TDM tile load: tile PW x `rows` fp32 starting at
// `gaddr`, written contiguously at LDS byte offset of `laddr`, iterated
// `niter` times with +SW floats global / +ASTRIDE floats LDS per iteration.
__device__ __forceinline__ void tdm_unfold(uint64_t gaddr, uint32_t laddr,
                                           int rows, int niter) {
    v4u g0;
    g0.x = 1u;                                          // count=1, user mode
    g0.y = laddr;                                       // LDS address (bytes)
    g0.z = (uint32_t)(gaddr & 0xFFFFFFFFu);             // global addr lo
    g0.w = (uint32_t)((gaddr >> 32) & 0x01FFFFFFu) | (2u << 30);  // hi | type=2

    v8i g1;
    g1[0] = (2 << 16) | (1 << 19);                      // data_size=4B | iterate
    g1[1] = (int)((XW & 0xFFFF) << 16);                 // tensor_dim0 lo16
    g1[2] = (int)((XW >> 16) | ((PH & 0xFFFF) << 16));  // dim0 hi | dim1 lo
    g1[3] = (int)(PW << 16);                            // dim1 hi(0) | tile_dim0
    g1[4] = (int)(rows & 0xFFFF);                       // tile_dim1 | tile_dim2=0
    g1[5] = XW;                                         // dim0_stride lo32
    g1[6] = 0;                                          // stride hi | dim1_stride lo
    g1[7] = 0;

    v4i g2;
    g2[0] = 0;                                          // tensor_dim2
    g2[1] = ASTRIDE;                                    // LDS += 1604 floats/iter
    g2[2] = SW;                                         // global += 12 floats/iter
    g2[3] = (int)(((niter - 1) & 0xFFFF) << 16);        // gai hi=0 | iterate_count

    v4i g3 = (v4i){0, 0, 0, 0};

#if defined(__clang_major__) && (__clang_major__ >= 23)
    __builtin_amdgcn_tensor_load_to_lds(g0, g1, g2, g3,
                                        (v8i){0,0,0,0,0,0,0,0}, 0);
#else
    __builtin_amdgcn_tensor_load_to_lds(g0, g1, g2, g3, 0);
#endif
}

extern "C" __global__ __launch_bounds__(256)
void patcher_wmma_kernel(const float* __restrict__ x,
                         const float* __restrict__ w,
                         const float* __restrict__ bias,
                         float* __restrict__ out)
{
    extern __shared__ float ldsA[];       // [MTILE][ASTRIDE] = 205,312 bytes

    const int chunk  = blockIdx.x;        // 0..10 : patch chunk within batch
    const int batch  = blockIdx.y;        // 0..63
    const int p0     = chunk * MTILE;     // first patch of this block
    const int npatch = (NPATCH - p0 < MTILE) ? (NPATCH - p0) : MTILE;

    // ---- Stage A with two TDM descriptors (one wave issues; EXEC ignored).
    if (threadIdx.x < 32) {
        const uint64_t gaddr =
            (uint64_t)(uintptr_t)(x + (size_t)batch * (PH * XW) + (size_t)p0 * SW);
        const uint32_t laddr = (uint32_t)(uintptr_t)&ldsA[0];

        // chunk 0: x rows [0,32)  -> patch elements k in [0, 800)
        tdm_unfold(gaddr, laddr, ROWSPLIT, npatch);
        // chunk 1: x rows [32,64) -> patch elements k in [800, 1600)
        tdm_unfold(gaddr + (uint64_t)ROWSPLIT * XW * sizeof(float),
                   laddr + (uint32_t)(KHALF * sizeof(float)),
                   ROWSPLIT, npatch);
        // TDM completes in order: <=1 outstanding means chunk 0 has landed.
        __builtin_amdgcn_s_wait_tensorcnt(1);
    }
    __syncthreads();

    // ---- wave tiling: wave owns N columns [wave*32, wave*32+32)
    const int wave = (int)(threadIdx.x >> 5);
    const int lane = (int)(threadIdx.x & 31);
    const int l16  = lane & 15;
    const int hi   = lane >> 4;               // lane group: K offset 0 or 2

    const int c0 = wave * 32 + l16;           // N col for B-frag / C-tile 0
    const int c1 = c0 + 16;                   // N col for B-frag / C-tile 1

    // A fragment LDS offsets (floats): lane<16 holds {K,K+1} of patch l16,
    // lane>=16 holds {K+2,K+3} of patch l16 (ISA 32-bit A 16x4 layout).
    const int pa0 = l16 * ASTRIDE + hi * 2;           // M-subtile 0
    const int pa1 = (16 + l16) * ASTRIDE + hi * 2;    // M-subtile 1

    // B fragment rows: VGPR0 = row k (+2 for hi lanes), VGPR1 = row k+1 (+2).
    const float* wrow = w + (size_t)(hi * 2) * NDIM;

    v8f acc00 = (v8f){0,0,0,0,0,0,0,0};
    v8f acc01 = (v8f){0,0,0,0,0,0,0,0};
    v8f acc10 = (v8f){0,0,0,0,0,0,0,0};
    v8f acc11 = (v8f){0,0,0,0,0,0,0,0};

    // ---- K-half 0 (k in [0,800)) while TDM streams K-half 1 into LDS
#pragma unroll 4
    for (int k = 0; k < KHALF; k += 4) {
        v2f a0 = *(const v2f*)(ldsA + pa0 + k);       // ds_load_b64, no conflicts
        v2f a1 = *(const v2f*)(ldsA + pa1 + k);
        const float* wk = wrow + (size_t)k * NDIM;
        v2f b0, b1;
        b0.x = wk[c0];        b0.y = wk[NDIM + c0];
        b1.x = wk[c1];        b1.y = wk[NDIM + c1];

        acc00 = __builtin_amdgcn_wmma_f32_16x16x4_f32(false, a0, false, b0,
                                                      (short)0, acc00, false, false);
        acc01 = __builtin_amdgcn_wmma_f32_16x16x4_f32(false, a0, false, b1,
                                                      (short)0, acc01, false, false);
        acc10 = __builtin_amdgcn_wmma_f32_16x16x4_f32(false, a1, false, b0,
                                                      (short)0, acc10, false, false);
        acc11 = __builtin_amdgcn_wmma_f32_16x16x4_f32(false, a1, false, b1,
                                                      (short)0, acc11, false, false);
    }

    // ---- wait for K-half 1 DMA, then finish the reduction
    if (threadIdx.x < 32) {
        __builtin_amdgcn_s_wait_tensorcnt(0);
    }
    __syncthreads();

#pragma unroll 4
    for (int k = KHALF; k < KDIM; k += 4) {
        v2f a0 = *(const v2f*)(ldsA + pa0 + k);
        v2f a1 = *(const v2f*)(ldsA + pa1 + k);
        const float* wk = wrow + (size_t)k * NDIM;
        v2f b0, b1;
        b0.x = wk[c0];        b0.y = wk[NDIM + c0];
        b1.x = wk[c1];        b1.y = wk[NDIM + c1];

        acc00 = __builtin_amdgcn_wmma_f32_16x16x4_f32(false, a0, false, b0,
                                                      (short)0, acc00, false, false);
        acc01 = __builtin_amdgcn_wmma_f32_16x16x4_f32(false, a0, false, b1,
                                                      (short)0, acc01, false, false);
        acc10 = __builtin_amdgcn_wmma_f32_16x16x4_f32(false, a1, false, b0,
                                                      (short)0, acc10, false, false);
        acc11 = __builtin_amdgcn_wmma_f32_16x16x4_f32(false, a1, false, b1,
                                                      (short)0, acc11, false, false);
    }

    // ---- epilogue: bias + exact GELU; C/D layout: VGPR r -> M = r (+8 hi lanes)
    const float bias0 = bias[c0];
    const float bias1 = bias[c1];

#pragma unroll
    for (int r = 0; r < 8; ++r) {
        const int prow0 = r + hi * 8;            // patch-in-block, M-subtile 0
        const int prow1 = 16 + prow0;            // M-subtile 1
        const int np0 = p0 + prow0;
        const int np1 = p0 + prow1;
        if (np0 < NPATCH) {
            const size_t o = ((size_t)batch * NPATCH + np0) * NDIM;
            out[o + c0] = gelu_exact(acc00[r] + bias0);
            out[o + c1] = gelu_exact(acc01[r] + bias1);
        }
        if (np1 < NPATCH) {
            const size_t o = ((size_t)batch * NPATCH + np1) * NDIM;
            out[o + c0] = gelu_exact(acc10[r] + bias0);
            out[o + c1] = gelu_exact(acc11[r] + bias1);
        }
    }
}

extern "C" void kernel_launch(void* const* d_in, const int* in_sizes, int n_in,
                              void* d_out, int out_size, void* d_ws, size_t ws_size,
                              hipStream_t stream) {
    (void)in_sizes; (void)n_in; (void)d_ws; (void)ws_size; (void)out_size;
    const float* x    = (const float*)d_in[0];   // [64,1,64,4000] fp32
    const float* w    = (const float*)d_in[1];   // [1600,256] fp32
    const float* bias = (const float*)d_in[2];   // [256] fp32
    float* out        = (float*)d_out;           // [64,332,256] fp32

    dim3 grid((NPATCH + MTILE - 1) / MTILE, NBATCH);   // 11 x 64
    dim3 block(256);                                    // 8 waves (wave32)
    size_t shmem = (size_t)MTILE * ASTRIDE * sizeof(float);  // 205,312 B
    hipLaunchKernelGGL(patcher_wmma_kernel, grid, block, shmem, stream,
                       x, w, bias, out);
}